// Sinkhorn_custom_29085518528514
// MI455X (gfx1250) — compile-verified
//
#include <hip/hip_runtime.h>
#include <math.h>

// ---------------- problem constants (fixed by reference setup) ----------------
constexpr int kBS = 32;
constexpr int kN  = 2048;
constexpr int kM  = 512;
constexpr float kEpsInv = 10.0f;          // 1/EPSILON
constexpr float kMu = 1.0f / 2048.0f;     // uniform row marginal
constexpr float kNu = 1.0f / 512.0f;      // uniform col marginal
constexpr int kIters = 200;

constexpr int RBLK  = 128;                // rows per workgroup tile
constexpr int NBLK  = kN / RBLK;          // 16 row-blocks per batch
constexpr int PITCH = kM + 4;             // 516: LDS row pitch (bank de-conflict, 16B aligned)
constexpr int TPB   = 256;                // 8 waves (wave32)

typedef __attribute__((ext_vector_type(2))) float v2f;
typedef __attribute__((ext_vector_type(4))) float v4f;
typedef __attribute__((ext_vector_type(8))) float v8f;

// smem layout (floats): tile[RBLK][PITCH] | v[kM] | u[RBLK]
constexpr int SMEM_FLOATS = RBLK * PITCH + kM + RBLK;
constexpr int SMEM_BYTES  = SMEM_FLOATS * 4;   // 266,752 B  (<= 320KB WGP LDS)

// ---------------- kernel 0: G = exp(-C/eps), stored into d_out ----------------
__global__ __launch_bounds__(TPB) void k_expG(const float* __restrict__ C,
                                              float* __restrict__ G, size_t n4) {
  size_t i = (size_t)blockIdx.x * blockDim.x + threadIdx.x;
  if (i >= n4) return;
  v4f c = *(const v4f*)(C + i * 4);
  v4f g;
  g.x = expf(-kEpsInv * c.x);
  g.y = expf(-kEpsInv * c.y);
  g.z = expf(-kEpsInv * c.z);
  g.w = expf(-kEpsInv * c.w);
  *(v4f*)(G + i * 4) = g;
}

// ---------------- kernel 0b: init partial col-sums so first v = 1/m -----------
__global__ __launch_bounds__(TPB) void k_initP(float* __restrict__ p, int n) {
  int i = blockIdx.x * blockDim.x + threadIdx.x;
  if (i < n) p[i] = 1.0f / (float)NBLK;   // sum over NBLK blocks = 1 -> v = nu/1 = 1/m
}

// ---------------- one fused Sinkhorn iteration ---------------------------------
// Each WG owns a 128x512 fp32 tile of one batch's G (L2-resident), staged in LDS
// once via GLOBAL_LOAD_ASYNC_TO_LDS_B128 and used by both reductions. Both
// reductions run on V_WMMA_F32_16X16X4_F32 with G in the B slot and the vector
// broadcast into all A rows.
__global__ __launch_bounds__(TPB) void k_iter(const float* __restrict__ G,
                                              const float* __restrict__ pprev,
                                              float* __restrict__ pnext,
                                              float* __restrict__ ubuf) {
  extern __shared__ float smem[];
  float* tile = smem;                          // [RBLK][PITCH]  (LDS offset 0)
  float* vsh  = smem + RBLK * PITCH;           // [kM]
  float* ush  = vsh + kM;                      // [RBLK]

  const int b    = blockIdx.x / NBLK;
  const int blk  = blockIdx.x % NBLK;
  const int tid  = threadIdx.x;
  const int lane = tid & 31;
  const int wv   = tid >> 5;                   // wave 0..7
  const int lm   = lane & 15;                  // N index within fragment
  const int lh   = lane >> 4;                  // K half select (0 or 1)

  // Kick off the async tile copy first so it overlaps the v[j] reduction below.
  // No static LDS in this kernel -> dynamic smem[] begins at LDS byte offset 0,
  // so the padded-pitch destination offset can be computed arithmetically.
  const float* gbase = G + ((size_t)b * kN + (size_t)blk * RBLK) * kM;
  for (int idx = tid * 4; idx < RBLK * kM; idx += TPB * 4) {
    int r = idx >> 9;          // /512
    int c = idx & (kM - 1);
    unsigned lds_off = (unsigned)((r * PITCH + c) * 4);
    unsigned long long gaddr = (unsigned long long)(const void*)(gbase + idx);
    asm volatile("global_load_async_to_lds_b128 %0, %1, off"
                 :: "v"(lds_off), "v"(gaddr) : "memory");
  }

  // v[j] = nu / sum_w pprev[b][w][j]   (redundant per WG; 8K loads, cheap)
  for (int j = tid; j < kM; j += TPB) {
    const float* p = pprev + (size_t)b * NBLK * kM + j;
    float s = 0.0f;
#pragma unroll
    for (int w = 0; w < NBLK; ++w) s += p[w * kM];
    vsh[j] = kNu / s;
  }

  // Drain this wave's async copies, then make the tile visible to all waves.
  asm volatile("s_wait_asynccnt 0" ::: "memory");
  __syncthreads();

  // ---- Phase A: row sums. Wave wv owns rows [16*wv, 16*wv+16).
  // C[i][j'] = sum_k v[k0+k] * G[r0+j'][k0+k]  -> every row of C equals r[r0+j'].
  {
    const int r0 = wv * 16;
    v8f acc0 = {};
    v8f acc1 = {};
    for (int k0 = 0; k0 < kM; k0 += 8) {
      v2f a0 = *(const v2f*)&vsh[k0 + 2 * lh];
      v2f b0 = *(const v2f*)&tile[(r0 + lm) * PITCH + k0 + 2 * lh];
      acc0 = __builtin_amdgcn_wmma_f32_16x16x4_f32(false, a0, false, b0,
                                                   (short)0, acc0, false, false);
      v2f a1 = *(const v2f*)&vsh[k0 + 4 + 2 * lh];
      v2f b1 = *(const v2f*)&tile[(r0 + lm) * PITCH + k0 + 4 + 2 * lh];
      acc1 = __builtin_amdgcn_wmma_f32_16x16x4_f32(false, a1, false, b1,
                                                   (short)0, acc1, false, false);
    }
    float r = acc0[0] + acc1[0];          // row sum for row r0+lm (all lanes valid)
    float uval = kMu / r;
    if (lane < 16) {
      ush[r0 + lm] = uval;
      ubuf[(size_t)b * kN + (size_t)blk * RBLK + r0 + lm] = uval;
    }
  }
  __syncthreads();

  // ---- Phase B: this block's contribution to column sums.
  // C[i][j'] = sum_k u[k0+k] * G[k0+k][j0+j']  (k runs over the 128 tile rows)
  for (int q = 0; q < 4; ++q) {
    const int j0 = (wv + 8 * q) * 16;     // 8 waves x 4 groups = 32 groups = 512 cols
    v8f acc0 = {};
    v8f acc1 = {};
    for (int k0 = 0; k0 < RBLK; k0 += 8) {
      v2f a0 = *(const v2f*)&ush[k0 + 2 * lh];
      v2f b0;
      b0.x = tile[(k0 + 2 * lh) * PITCH + j0 + lm];
      b0.y = tile[(k0 + 2 * lh + 1) * PITCH + j0 + lm];
      acc0 = __builtin_amdgcn_wmma_f32_16x16x4_f32(false, a0, false, b0,
                                                   (short)0, acc0, false, false);
      v2f a1 = *(const v2f*)&ush[k0 + 4 + 2 * lh];
      v2f b1;
      b1.x = tile[(k0 + 4 + 2 * lh) * PITCH + j0 + lm];
      b1.y = tile[(k0 + 5 + 2 * lh) * PITCH + j0 + lm];
      acc1 = __builtin_amdgcn_wmma_f32_16x16x4_f32(false, a1, false, b1,
                                                   (short)0, acc1, false, false);
    }
    if (lane < 16) {
      pnext[((size_t)b * NBLK + blk) * kM + j0 + lm] = acc0[0] + acc1[0];
    }
  }
}

// ---------------- final: P = u * G * v, rescaling G in place in d_out ----------
__global__ __launch_bounds__(TPB) void k_final(float* __restrict__ P,
                                               const float* __restrict__ plast,
                                               const float* __restrict__ ubuf) {
  __shared__ float vsh[kM];
  const int b   = blockIdx.x / NBLK;
  const int blk = blockIdx.x % NBLK;
  const int tid = threadIdx.x;
  for (int j = tid; j < kM; j += TPB) {
    const float* p = plast + (size_t)b * NBLK * kM + j;
    float s = 0.0f;
#pragma unroll
    for (int w = 0; w < NBLK; ++w) s += p[w * kM];
    vsh[j] = kNu / s;
  }
  __syncthreads();
  float* base = P + ((size_t)b * kN + (size_t)blk * RBLK) * kM;
  const float* ub = ubuf + (size_t)b * kN + (size_t)blk * RBLK;
  for (int idx = tid * 4; idx < RBLK * kM; idx += TPB * 4) {
    int r = idx >> 9;
    int c = idx & (kM - 1);
    v4f g = *(const v4f*)(base + idx);
    float u = ub[r];
    v4f o;
    o.x = u * g.x * vsh[c + 0];
    o.y = u * g.y * vsh[c + 1];
    o.z = u * g.z * vsh[c + 2];
    o.w = u * g.w * vsh[c + 3];
    *(v4f*)(base + idx) = o;
  }
}

// ------------------------------- launcher --------------------------------------
extern "C" void kernel_launch(void* const* d_in, const int* in_sizes, int n_in,
                              void* d_out, int out_size, void* d_ws, size_t ws_size,
                              hipStream_t stream) {
  const float* C = (const float*)d_in[0];    // [32][2048][512] fp32
  float* out = (float*)d_out;                // holds G during iterations, P at end

  // workspace: two ping-pong partial-column-sum buffers + u vector (~2.25 MB)
  float* pA = (float*)d_ws;                            // [kBS][NBLK][kM]
  float* pB = pA + (size_t)kBS * NBLK * kM;
  float* ub = pB + (size_t)kBS * NBLK * kM;            // [kBS][kN]

  // allow >64KB dynamic LDS for the iteration kernel (no-op if already set)
  (void)hipFuncSetAttribute((const void*)k_iter,
                            hipFuncAttributeMaxDynamicSharedMemorySize, SMEM_BYTES);

  const size_t total = (size_t)kBS * kN * kM;          // 33,554,432
  const size_t n4 = total / 4;
  k_expG<<<(unsigned)((n4 + TPB - 1) / TPB), TPB, 0, stream>>>(C, out, n4);

  const int np = kBS * NBLK * kM;
  k_initP<<<(np + TPB - 1) / TPB, TPB, 0, stream>>>(pA, np);

  for (int t = 0; t < kIters; ++t) {
    const float* pp = (t & 1) ? pB : pA;
    float*       pn = (t & 1) ? pA : pB;
    k_iter<<<kBS * NBLK, TPB, SMEM_BYTES, stream>>>(out, pp, pn, ub);
  }
  // kIters even -> last write landed in pA
  k_final<<<kBS * NBLK, TPB, 0, stream>>>(out, pA, ub);
}